// GAT2_6408091205707
// MI455X (gfx1250) — compile-verified
//
#include <hip/hip_runtime.h>

typedef __attribute__((ext_vector_type(2))) float v2f;
typedef __attribute__((ext_vector_type(8))) float v8f;

#define DIM 16
#define NEG_SLOPE 0.2f

__device__ __forceinline__ float leaky(float e) { return e >= 0.f ? e : NEG_SLOPE * e; }

// Monotone order-preserving encode of float into uint (for atomicMax-based segment max).
__device__ __forceinline__ unsigned fenc(float f) {
  unsigned u = __float_as_uint(f);
  return (u & 0x80000000u) ? ~u : (u | 0x80000000u);
}
__device__ __forceinline__ float fdec(unsigned u) {
  return __uint_as_float((u & 0x80000000u) ? (u & 0x7FFFFFFFu) : ~u);
}

// Zero-pad X[n,14] -> XP[n,16]
__global__ void gat_pad_x(const float* __restrict__ X, float* __restrict__ XP, int n) {
  int idx = blockIdx.x * blockDim.x + threadIdx.x;
  if (idx >= n * DIM) return;
  int i = idx >> 4, c = idx & 15;
  XP[idx] = (c < 14) ? X[i * 14 + c] : 0.f;
}

// Zero-pad W1[14,16] -> WP[16,16]
__global__ void gat_pad_w(const float* __restrict__ W, float* __restrict__ WP) {
  int idx = threadIdx.x;  // 256 threads
  int r = idx >> 4, c = idx & 15;
  WP[idx] = (r < 14) ? W[r * DIM + c] : 0.f;
}

// ---------------------------------------------------------------------------
// H[n,16] = X16[n,16] @ W16[16,16] using V_WMMA_F32_16X16X4_F32 (4 k-steps).
// One wave per 16-row tile; inputs pre-padded so all loads are unconditional.
// A: 16x4 f32 (lanes 0-15: M=lane, v0/v1 = K0/K1; lanes 16-31: same M, K2/K3).
// B: 4x16 f32 (v0: lanes0-15 = row K0, lanes16-31 = row K2; v1: K1/K3).
// C/D: 8 VGPRs, element (M = r + 8*half, N = lane&15).
// All addressing 32-bit; store bound check is wave-uniform (full-tile fast path).
// ---------------------------------------------------------------------------
__global__ void gat_proj_wmma(const float* __restrict__ X16, const float* __restrict__ W16,
                              float* __restrict__ H, int n) {
  int wave = (int)((blockIdx.x * blockDim.x + threadIdx.x) >> 5);
  int lane = threadIdx.x & 31;
  int ntiles = (n + 15) >> 4;
  if (wave >= ntiles) return;  // wave-uniform: EXEC all-1s at the WMMAs
  int half = lane >> 4;
  int l = lane & 15;
  int rowA = wave * 16 + l;
  if (rowA >= n) rowA = n - 1;  // clamp (dup rows never stored)
  v8f c = {};
#pragma unroll
  for (int k0 = 0; k0 < DIM; k0 += 4) {
    int ka = k0 + half * 2;
    float2 av = *(const float2*)&X16[rowA * DIM + ka];  // 8B-aligned b64 load
    v2f a, b;
    a.x = av.x;
    a.y = av.y;
    b.x = W16[ka * DIM + l];
    b.y = W16[(ka + 1) * DIM + l];
    c = __builtin_amdgcn_wmma_f32_16x16x4_f32(false, a, false, b, (short)0, c,
                                              false, false);
  }
  int base = (wave * 16 + half * 8) * DIM + l;
  if (wave * 16 + 16 <= n) {  // wave-uniform fast path: full tile
#pragma unroll
    for (int r = 0; r < 8; ++r) H[base + r * DIM] = c[r];
  } else {  // partial last tile (never taken for n % 16 == 0)
#pragma unroll
    for (int r = 0; r < 8; ++r) {
      int row = wave * 16 + r + half * 8;
      if (row < n) H[row * DIM + l] = c[r];
    }
  }
}

// as[i] = h[i] . a_src ; ad[i] = h[i] . a_dst
__global__ void gat_alphas(const float* __restrict__ H, const float* __restrict__ a_src,
                           const float* __restrict__ a_dst, float* __restrict__ AS,
                           float* __restrict__ AD, int n) {
  int i = blockIdx.x * blockDim.x + threadIdx.x;
  if (i >= n) return;
  float s = 0.f, d = 0.f;
#pragma unroll
  for (int j = 0; j < 4; ++j) {
    float4 h = *(const float4*)&H[i * DIM + j * 4];
    s += h.x * a_src[j * 4 + 0] + h.y * a_src[j * 4 + 1] +
         h.z * a_src[j * 4 + 2] + h.w * a_src[j * 4 + 3];
    d += h.x * a_dst[j * 4 + 0] + h.y * a_dst[j * 4 + 1] +
         h.z * a_dst[j * 4 + 2] + h.w * a_dst[j * 4 + 3];
  }
  AS[i] = s;
  AD[i] = d;
}

// Seed segment-max with the self-loop edge (no atomics needed).
__global__ void gat_self_init(const float* __restrict__ AS, const float* __restrict__ AD,
                              unsigned* __restrict__ MK, int n) {
  int i = blockIdx.x * blockDim.x + threadIdx.x;
  if (i >= n) return;
  MK[i] = fenc(leaky(AS[i] + AD[i]));
}

// Segment max over explicit edges; 4 edges/thread via int4 index loads.
__global__ void gat_edge_max(const int* __restrict__ src, const int* __restrict__ dst,
                             const float* __restrict__ AS, const float* __restrict__ AD,
                             unsigned* __restrict__ MK, int E) {
  int tid = blockIdx.x * blockDim.x + threadIdx.x;
  int stride = gridDim.x * blockDim.x;
  int E4 = E >> 2;
  const int4* s4p = (const int4*)src;
  const int4* d4p = (const int4*)dst;
  for (int q = tid; q < E4; q += stride) {
    int4 s4 = s4p[q], d4 = d4p[q];
    atomicMax(&MK[d4.x], fenc(leaky(AS[s4.x] + AD[d4.x])));
    atomicMax(&MK[d4.y], fenc(leaky(AS[s4.y] + AD[d4.y])));
    atomicMax(&MK[d4.z], fenc(leaky(AS[s4.z] + AD[d4.z])));
    atomicMax(&MK[d4.w], fenc(leaky(AS[s4.w] + AD[d4.w])));
  }
  if (tid < (E & 3)) {
    int e = (E & ~3) + tid;
    int s = src[e], d = dst[e];
    atomicMax(&MK[d], fenc(leaky(AS[s] + AD[d])));
  }
}

// Seed denom + numerator with the self-loop contribution (plain stores).
__global__ void gat_node_seed(const float* __restrict__ H, const float* __restrict__ AS,
                              const float* __restrict__ AD, const unsigned* __restrict__ MK,
                              float* __restrict__ DEN, float* __restrict__ S, int n) {
  int i = blockIdx.x * blockDim.x + threadIdx.x;
  if (i >= n) return;
  float m = fdec(MK[i]);
  float ex = expf(leaky(AS[i] + AD[i]) - m);
  DEN[i] = ex;
#pragma unroll
  for (int j = 0; j < 4; ++j) {
    float4 h = *(const float4*)&H[i * DIM + j * 4];
    float4 o = {ex * h.x, ex * h.y, ex * h.z, ex * h.w};
    *(float4*)&S[i * DIM + j * 4] = o;
  }
}

__device__ __forceinline__ void gat_edge_one(int s, int d, const float* __restrict__ H,
                                             const float* __restrict__ AS,
                                             const float* __restrict__ AD,
                                             const unsigned* __restrict__ MK,
                                             float* __restrict__ DEN,
                                             float* __restrict__ S) {
  float ex = expf(leaky(AS[s] + AD[d]) - fdec(MK[d]));
  atomicAdd(&DEN[d], ex);
  const float4* h4 = (const float4*)&H[s * DIM];
  float* sd = &S[d * DIM];
#pragma unroll
  for (int j = 0; j < 4; ++j) {
    float4 v = h4[j];
    atomicAdd(&sd[j * 4 + 0], ex * v.x);
    atomicAdd(&sd[j * 4 + 1], ex * v.y);
    atomicAdd(&sd[j * 4 + 2], ex * v.z);
    atomicAdd(&sd[j * 4 + 3], ex * v.w);
  }
}

// Heavy pass: per edge, accumulate denom and weighted h[src] into S[dst].
__global__ void gat_edge_acc(const int* __restrict__ src, const int* __restrict__ dst,
                             const float* __restrict__ H, const float* __restrict__ AS,
                             const float* __restrict__ AD, const unsigned* __restrict__ MK,
                             float* __restrict__ DEN, float* __restrict__ S, int E) {
  int tid = blockIdx.x * blockDim.x + threadIdx.x;
  int stride = gridDim.x * blockDim.x;
  int E4 = E >> 2;
  const int4* s4p = (const int4*)src;
  const int4* d4p = (const int4*)dst;
  for (int q = tid; q < E4; q += stride) {
    __builtin_prefetch(&s4p[q + 1024], 0, 1);  // global_prefetch_b8 on edge stream
    __builtin_prefetch(&d4p[q + 1024], 0, 1);
    int4 s4 = s4p[q], d4 = d4p[q];
    gat_edge_one(s4.x, d4.x, H, AS, AD, MK, DEN, S);
    gat_edge_one(s4.y, d4.y, H, AS, AD, MK, DEN, S);
    gat_edge_one(s4.z, d4.z, H, AS, AD, MK, DEN, S);
    gat_edge_one(s4.w, d4.w, H, AS, AD, MK, DEN, S);
  }
  if (tid < (E & 3)) {
    int e = (E & ~3) + tid;
    gat_edge_one(src[e], dst[e], H, AS, AD, MK, DEN, S);
  }
}

// out[i,c] = S[i,c]/DEN[i] + b[c] ; optional ReLU.
__global__ void gat_node_fin(const float* __restrict__ S, const float* __restrict__ DEN,
                             const float* __restrict__ b, float* __restrict__ OUT, int n,
                             int do_relu) {
  int idx = blockIdx.x * blockDim.x + threadIdx.x;
  if (idx >= n * DIM) return;
  int i = idx >> 4, cidx = idx & 15;
  float v = S[idx] / DEN[i] + b[cidx];
  if (do_relu) v = v > 0.f ? v : 0.f;
  OUT[idx] = v;
}

extern "C" void kernel_launch(void* const* d_in, const int* in_sizes, int n_in,
                              void* d_out, int out_size, void* d_ws, size_t ws_size,
                              hipStream_t stream) {
  const float* x = (const float*)d_in[0];
  const int* ei = (const int*)d_in[1];
  const float* W1 = (const float*)d_in[2];
  const float* as1 = (const float*)d_in[3];
  const float* ad1 = (const float*)d_in[4];
  const float* b1 = (const float*)d_in[5];
  const float* W2 = (const float*)d_in[6];
  const float* as2 = (const float*)d_in[7];
  const float* ad2 = (const float*)d_in[8];
  const float* b2 = (const float*)d_in[9];
  float* out = (float*)d_out;

  const int NFEAT = 14;
  int n = in_sizes[0] / NFEAT;  // 100000
  int E = in_sizes[1] / 2;      // 3200000
  const int* src = ei;
  const int* dst = ei + E;

  // Workspace layout: 68*n + 256 floats (~27.3 MB for n=100000).
  float* fws = (float*)d_ws;
  float* H = fws;                       // n*16  (h of current layer)
  float* S = H + (size_t)n * DIM;       // n*16  (unnormalized numerator)
  float* O1 = S + (size_t)n * DIM;      // n*16  (layer-1 output after relu)
  float* XP = O1 + (size_t)n * DIM;     // n*16  (zero-padded input features)
  float* AS = XP + (size_t)n * DIM;     // n
  float* AD = AS + n;                   // n
  float* DEN = AD + n;                  // n
  unsigned* MK = (unsigned*)(DEN + n);  // n
  float* WP = (float*)(MK + n);         // 256   (zero-padded W1)

  const int TB = 256;
  int ntiles = (n + 15) >> 4;
  dim3 gProj((ntiles + 7) / 8);  // 8 wave32 waves per 256-thread block
  dim3 gNode((n + TB - 1) / TB);
  dim3 gElem((n * DIM + TB - 1) / TB);
  dim3 gEdge(((E >> 2) + TB - 1) / TB);

  // ---------------- Layer 1 ----------------
  gat_pad_x<<<gElem, TB, 0, stream>>>(x, XP, n);
  gat_pad_w<<<1, 256, 0, stream>>>(W1, WP);
  gat_proj_wmma<<<gProj, TB, 0, stream>>>(XP, WP, H, n);
  gat_alphas<<<gNode, TB, 0, stream>>>(H, as1, ad1, AS, AD, n);
  gat_self_init<<<gNode, TB, 0, stream>>>(AS, AD, MK, n);
  gat_edge_max<<<gEdge, TB, 0, stream>>>(src, dst, AS, AD, MK, E);
  gat_node_seed<<<gNode, TB, 0, stream>>>(H, AS, AD, MK, DEN, S, n);
  gat_edge_acc<<<gEdge, TB, 0, stream>>>(src, dst, H, AS, AD, MK, DEN, S, E);
  gat_node_fin<<<gElem, TB, 0, stream>>>(S, DEN, b1, O1, n, 1);

  // ---------------- Layer 2 ----------------
  gat_proj_wmma<<<gProj, TB, 0, stream>>>(O1, W2, H, n);
  gat_alphas<<<gNode, TB, 0, stream>>>(H, as2, ad2, AS, AD, n);
  gat_self_init<<<gNode, TB, 0, stream>>>(AS, AD, MK, n);
  gat_edge_max<<<gEdge, TB, 0, stream>>>(src, dst, AS, AD, MK, E);
  gat_node_seed<<<gNode, TB, 0, stream>>>(H, AS, AD, MK, DEN, S, n);
  gat_edge_acc<<<gEdge, TB, 0, stream>>>(src, dst, H, AS, AD, MK, DEN, S, E);
  gat_node_fin<<<gElem, TB, 0, stream>>>(S, DEN, b2, out, n, 0);
}